// SmearImages_70549132804589
// MI455X (gfx1250) — compile-verified
//
#include <hip/hip_runtime.h>
#include <cmath>

typedef __attribute__((ext_vector_type(2))) float v2f;
typedef __attribute__((ext_vector_type(8))) float v8f;

// One block = one camera (b,ip) x 256 voxels.
// Phase A: hom = M(3x4) @ [x,y,z,1] via V_WMMA_F32_16X16X4_F32, branchless
//          operand construction (unconditional loads + selects, EXEC stays full).
// Phase B: per-thread perspective divide, branchless bilinear gather (C=3),
//          view direction, 8 coalesced channel stores.
__global__ __launch_bounds__(256)
void SmearImages_kernel(const float* __restrict__ images,   // (B, IP, C, H, W)
                        const float* __restrict__ transf,   // (B, IP, 3, 4)
                        const float* __restrict__ T_cw,     // (B, IP, 4, 4)
                        const float* __restrict__ coords,   // (B, 3, N)
                        float* __restrict__ out,            // (B, IP, 8, N)
                        int IP, int C, int H, int W, int N)
{
    __shared__ float homLDS[3][256];

    const int tilesPerCam = N >> 8;              // N / 256
    const int tile = blockIdx.x % tilesPerCam;
    const int cam  = blockIdx.x / tilesPerCam;   // cam in [0, B*IP)
    const int b    = cam / IP;

    const int tid   = threadIdx.x;
    const int lane  = tid & 31;
    const int wave  = tid >> 5;                  // 8 waves
    const int nbase = tile << 8;

    const float* __restrict__ M  = transf + (size_t)cam * 12;   // 3x4 row-major
    const float* __restrict__ cx = coords + (size_t)b * 3 * N;

    // ---- Phase A: WMMA projection (branchless operands) ----------------
    // A (16x4 f32): lanes 0-15 -> row=lane, {K0,K1}; lanes 16-31 -> row=lane-16, {K2,K3}.
    const int  half    = lane >> 4;              // 0 or 1
    const int  arow    = lane & 15;
    const bool rowLive = (arow < 3);
    const int  aidx    = rowLive ? (arow * 4 + half * 2) : 0;  // clamp addr in-bounds
    const float aL0 = M[aidx];
    const float aL1 = M[aidx + 1];
    v2f a;
    a.x = rowLive ? aL0 : 0.0f;
    a.y = rowLive ? aL1 : 0.0f;

    // B (4x16 f32): lanes 0-15 hold {row0=x, row1=y} of point N=lane;
    //               lanes 16-31 hold {row2=z, row3=1} of point N=lane-16.
    const int pb = nbase + wave * 32 + arow;
    const float bx0 = cx[(half * 2) * N + pb];        // x or z
    const float by0 = cx[N + pb];                     // y (unused for half=1)
    const float bx1 = cx[(half * 2) * N + pb + 16];
    const float by1 = cx[N + pb + 16];
    v2f b0, b1;
    b0.x = bx0;  b0.y = half ? 1.0f : by0;
    b1.x = bx1;  b1.y = half ? 1.0f : by1;

    const v8f czero = {};
    v8f d0 = __builtin_amdgcn_wmma_f32_16x16x4_f32(false, a, false, b0,
                                                   (short)0, czero, false, false);
    v8f d1 = __builtin_amdgcn_wmma_f32_16x16x4_f32(false, a, false, b1,
                                                   (short)0, czero, false, false);

    // D (16x16 f32): lane n (0-15), VGPR r -> row r, point n. Rows 0..2 = hom.
    if (lane < 16) {
        const int l0 = wave * 32 + lane;
        homLDS[0][l0]      = d0[0]; homLDS[1][l0]      = d0[1]; homLDS[2][l0]      = d0[2];
        homLDS[0][l0 + 16] = d1[0]; homLDS[1][l0 + 16] = d1[1]; homLDS[2][l0 + 16] = d1[2];
    }
    __syncthreads();

    // ---- Phase B: per-point sampling (branchless) ----------------------
    const int n = nbase + tid;
    const float hx = homLDS[0][tid];
    const float hy = homLDS[1][tid];
    const float hz = homLDS[2][tid];

    const float depth = hz;
    const float safe  = (fabsf(depth) > 1e-8f) ? depth : 1e-8f;
    const float u = hx / safe;
    const float v = hy / safe;
    const float valid = (depth > 0.0f && u >= 0.0f && u <= (float)(W - 1) &&
                         v >= 0.0f && v <= (float)(H - 1)) ? 1.0f : 0.0f;

    const float x0f = floorf(u), y0f = floorf(v);
    const float fx = u - x0f, fy = v - y0f;
    const int x0 = (int)x0f, y0 = (int)y0f;

    const float wgt[4] = { (1.0f - fx) * (1.0f - fy), fx * (1.0f - fy),
                           (1.0f - fx) * fy,          fx * fy };

    const int HW = H * W;
    const float* __restrict__ img = images + (size_t)cam * C * HW;
    float s0 = 0.0f, s1 = 0.0f, s2 = 0.0f;
#pragma unroll
    for (int t = 0; t < 4; ++t) {
        const int ix = x0 + (t & 1);
        const int iy = y0 + (t >> 1);
        const bool inb = (ix >= 0) & (ix < W) & (iy >= 0) & (iy < H);
        const int cix = min(max(ix, 0), W - 1);
        const int ciy = min(max(iy, 0), H - 1);
        const float w = inb ? wgt[t] : 0.0f;          // mask via weight, not EXEC
        const size_t base = (size_t)ciy * W + cix;    // always in-bounds
        s0 = fmaf(w, img[base],          s0);
        s1 = fmaf(w, img[HW + base],     s1);
        s2 = fmaf(w, img[2 * HW + base], s2);
    }

    // cam_center = -R^T t  (R = T[0:3,0:3], t = T[0:3,3])
    const float* __restrict__ T = T_cw + (size_t)cam * 16;
    const float ccx = -(T[0] * T[3] + T[4] * T[7] + T[8]  * T[11]);
    const float ccy = -(T[1] * T[3] + T[5] * T[7] + T[9]  * T[11]);
    const float ccz = -(T[2] * T[3] + T[6] * T[7] + T[10] * T[11]);

    const float px = cx[0 * N + n], py = cx[1 * N + n], pz = cx[2 * N + n];
    float vx = px - ccx, vy = py - ccy, vz = pz - ccz;
    const float nrm = sqrtf(vx * vx + vy * vy + vz * vz);
    const float inv = 1.0f / fmaxf(nrm, 1e-8f);
    vx *= inv; vy *= inv; vz *= inv;

    float* __restrict__ o = out + (size_t)cam * 8 * N;
    o[0 * N + n] = s0;
    o[1 * N + n] = s1;
    o[2 * N + n] = s2;
    o[3 * N + n] = depth;
    o[4 * N + n] = valid;
    o[5 * N + n] = vx;
    o[6 * N + n] = vy;
    o[7 * N + n] = vz;
}

extern "C" void kernel_launch(void* const* d_in, const int* in_sizes, int n_in,
                              void* d_out, int out_size, void* d_ws, size_t ws_size,
                              hipStream_t stream) {
    const float* images = (const float*)d_in[0];
    const float* transf = (const float*)d_in[1];
    const float* T_cw   = (const float*)d_in[2];
    const float* coords = (const float*)d_in[3];
    float* out = (float*)d_out;

    // Reference shapes: B=2, I=2, P=2, C=3, H=480, W=640, X=Y=Z=64.
    const int B = 2, C = 3, H = 480, W = 640;
    const int IP = in_sizes[2] / (16 * B);        // T_cw is (B, IP, 4, 4)
    const int N  = in_sizes[3] / (3 * B);         // coordinates is (B, 3, N)

    const int blocks = B * IP * (N / 256);
    SmearImages_kernel<<<blocks, 256, 0, stream>>>(images, transf, T_cw, coords,
                                                   out, IP, C, H, W, N);
}